// SoftPatternClassifier_33062658244960
// MI455X (gfx1250) — compile-verified
//
#include <hip/hip_runtime.h>
#include <hip/hip_bf16.h>

typedef float v2f __attribute__((ext_vector_type(2)));
typedef float v8f __attribute__((ext_vector_type(8)));

#define V_   8192   // vocab
#define DW   300    // D_WORD (GEMM K)
#define DD   3600   // D = P * NUM_DIAGS * L
#define BB   64
#define NN   256
#define PP   300
#define HIDN 256

// Fast sigmoid: v_mul + v_exp_f32 + v_add + v_rcp_f32 (no IEEE divide sequence).
__device__ __forceinline__ float fast_sigmoid(float x) {
    return __builtin_amdgcn_rcpf(1.0f + __expf(-x));
}

// ---------------------------------------------------------------------------
// Kernel 1: ts[v][d] = sigmoid( (diags @ emb)[d][v] + bias[d] )
// Computed as C' = A' x B' with A'[m][k] = emb[k][v0+m], B'[k][n] = diags[d0+n][k]
// so that C'[m][n] = (diags@emb)[d0+n][v0+m] and stores along d are coalesced.
// One wave per 16x16 output tile; 8 waves (8 v-tiles) per block.
// f32 WMMA layout (wave32): m = lane&15, h = lane>>4.
//   A (2 VGPRs): lane holds A'[m][2h], A'[m][2h+1]
//   B (2 VGPRs): lane holds B'[2h][n=m], B'[2h+1][n=m]   (contiguous in diags!)
//   C (8 VGPRs): acc[r] = C'[r + 8h][m]
// ---------------------------------------------------------------------------
__global__ __launch_bounds__(256) void ts_gemm_kernel(
    const float* __restrict__ emb,    // (300, 8192) row-major
    const float* __restrict__ diags,  // (3600, 300) row-major
    const float* __restrict__ bias,   // (3600,)
    float* __restrict__ ts)           // out (8192, 3600)
{
    const int lane = threadIdx.x & 31;
    const int wave = threadIdx.x >> 5;
    const int vt = blockIdx.x * 8 + wave;   // 0..511
    const int dt = blockIdx.y;              // 0..224
    const int v0 = vt * 16;
    const int d0 = dt * 16;
    const int m  = lane & 15;
    const int h  = lane >> 4;

    const float* aptr = emb   + (size_t)(2 * h) * V_ + (v0 + m);  // advance rows by k
    const float* bptr = diags + (size_t)(d0 + m) * DW + 2 * h;    // contiguous k pair

    v8f acc = {};
#if __has_builtin(__builtin_amdgcn_wmma_f32_16x16x4_f32)
    for (int k0 = 0; k0 < DW; k0 += 4) {
        v2f a, b;
        a.x = aptr[(size_t)k0 * V_];
        a.y = aptr[(size_t)(k0 + 1) * V_];
        b   = *(const v2f*)(bptr + k0);
        acc = __builtin_amdgcn_wmma_f32_16x16x4_f32(
            false, a, false, b, (short)0, acc, false, false);
    }
#else
    // scalar fallback: each lane computes its 8 C' elements directly
    for (int r = 0; r < 8; ++r) {
        float s = 0.f;
        const int vv = v0 + r + 8 * h;
        for (int k = 0; k < DW; ++k)
            s = fmaf(emb[(size_t)k * V_ + vv], diags[(size_t)(d0 + m) * DW + k], s);
        acc[r] = s;
    }
#endif

    const float bn = bias[d0 + m];
#pragma unroll
    for (int r = 0; r < 8; ++r) {
        const int vv = v0 + r + 8 * h;
        ts[(size_t)vv * DD + d0 + m] = fast_sigmoid(acc[r] + bn);
    }
}

// ---------------------------------------------------------------------------
// Kernel 2: the 6-state pattern recurrence. One (b, p) per lane.
// ts row slice for pattern p: offsets p*12 + [0..5] = diag0 (self loop),
// p*12 + 6 + [0..4] = diag1 (main transition). 48B aligned -> 3x float4.
// Double-buffered loads of the next token's transitions.
// ---------------------------------------------------------------------------
__global__ __launch_bounds__(320) void scan_kernel(
    const float* __restrict__ ts,      // (8192, 3600)
    const float* __restrict__ eps_raw, // (300, 5)
    const int*   __restrict__ docs,    // (64, 256)
    const int*   __restrict__ doc_lens,// (64,)
    float* __restrict__ scores)        // out (64, 300)
{
    const int b  = blockIdx.x;
    const int p  = threadIdx.x;
    const int pc = (p < PP) ? p : (PP - 1);   // clamp for safe OOB reads

    float e0, e1, e2, e3, e4;
    {
        const float* er = eps_raw + pc * 5;
        e0 = fast_sigmoid(er[0]);
        e1 = fast_sigmoid(er[1]);
        e2 = fast_sigmoid(er[2]);
        e3 = fast_sigmoid(er[3]);
        e4 = fast_sigmoid(er[4]);
    }
    const int n = doc_lens[b];                // < N by construction
    const int* dptr = docs + b * NN;
    const int eidx = pc / 100;                // end state = 3,4,5

    float h0 = 1.f, h1 = 0.f, h2 = 0.f, h3 = 0.f, h4 = 0.f, h5 = 0.f;
    float sc = 0.f;

    float4 A, Bq, C;
    if (n > 0) {
        const float* base = ts + (size_t)dptr[0] * DD + pc * 12;
        A  = *(const float4*)(base);
        Bq = *(const float4*)(base + 4);
        C  = *(const float4*)(base + 8);
    }
    for (int i = 0; i < n; ++i) {
        const float t0 = A.x,  t1 = A.y,  t2 = A.z,  t3 = A.w;
        const float t4 = Bq.x, t5 = Bq.y;
        const float u0 = Bq.z, u1 = Bq.w, u2 = C.x, u3 = C.y, u4 = C.z;
        if (i + 1 < n) {   // prefetch next step while computing this one
            const float* base = ts + (size_t)dptr[i + 1] * DD + pc * 12;
            A  = *(const float4*)(base);
            Bq = *(const float4*)(base + 4);
            C  = *(const float4*)(base + 8);
        }
        const float ae0 = h0;
        const float ae1 = fmaf(h0, e0, h1);
        const float ae2 = fmaf(h1, e1, h2);
        const float ae3 = fmaf(h2, e2, h3);
        const float ae4 = fmaf(h3, e3, h4);
        const float ae5 = fmaf(h4, e4, h5);
        h0 = fmaf(ae0, t0, 1.0f);
        h1 = fmaf(ae1, t1, ae0 * u0);
        h2 = fmaf(ae2, t2, ae1 * u1);
        h3 = fmaf(ae3, t3, ae2 * u2);
        h4 = fmaf(ae4, t4, ae3 * u3);
        h5 = fmaf(ae5, t5, ae4 * u4);
        sc += (eidx == 0) ? h3 : (eidx == 1) ? h4 : h5;
    }
    if (p < PP) scores[b * PP + p] = sc;
}

// ---------------------------------------------------------------------------
// Kernel 3: out = relu(scores @ w1 + b1) @ w2 + b2   (64x300x256x2 -> tiny)
// One block per batch row; LDS tree reduction for the 2 output classes.
// ---------------------------------------------------------------------------
__global__ __launch_bounds__(256) void mlp_kernel(
    const float* __restrict__ scores, // (64, 300)
    const float* __restrict__ w1,     // (300, 256)
    const float* __restrict__ b1,     // (256,)
    const float* __restrict__ w2,     // (256, 2)
    const float* __restrict__ b2,     // (2,)
    float* __restrict__ out)          // (64, 2)
{
    const int b = blockIdx.x;
    const int j = threadIdx.x;
    float acc = b1[j];
    const float* srow = scores + b * PP;
    for (int pq = 0; pq < PP; ++pq)
        acc = fmaf(srow[pq], w1[pq * HIDN + j], acc);
    const float hh = acc > 0.f ? acc : 0.f;

    __shared__ float r0[HIDN], r1[HIDN];
    r0[j] = hh * w2[j * 2 + 0];
    r1[j] = hh * w2[j * 2 + 1];
    __syncthreads();
    for (int s = HIDN / 2; s > 0; s >>= 1) {
        if (j < s) { r0[j] += r0[j + s]; r1[j] += r1[j + s]; }
        __syncthreads();
    }
    if (j == 0) {
        out[b * 2 + 0] = r0[0] + b2[0];
        out[b * 2 + 1] = r1[0] + b2[1];
    }
}

// ---------------------------------------------------------------------------
extern "C" void kernel_launch(void* const* d_in, const int* in_sizes, int n_in,
                              void* d_out, int out_size, void* d_ws, size_t ws_size,
                              hipStream_t stream) {
    const float* emb      = (const float*)d_in[0]; // (300, 8192)
    const float* diags    = (const float*)d_in[1]; // (3600, 300)
    const float* bias     = (const float*)d_in[2]; // (3600, 1)
    const float* epsilon  = (const float*)d_in[3]; // (300, 5)
    const float* w1       = (const float*)d_in[4]; // (300, 256)
    const float* b1       = (const float*)d_in[5]; // (256,)
    const float* w2       = (const float*)d_in[6]; // (256, 2)
    const float* b2       = (const float*)d_in[7]; // (2,)
    const int*   docs     = (const int*)d_in[8];   // (64, 256)
    const int*   doc_lens = (const int*)d_in[9];   // (64,)

    // Workspace layout: ts (8192*3600 f32 = 118 MB), then scores (64*300 f32).
    float* ts     = (float*)d_ws;
    float* scores = ts + (size_t)V_ * DD;

    dim3 g1(64, 225);   // 64*8 = 512 v-tiles, 225 d-tiles
    ts_gemm_kernel<<<g1, 256, 0, stream>>>(emb, diags, bias, ts);

    scan_kernel<<<BB, 320, 0, stream>>>(ts, epsilon, docs, doc_lens, scores);

    mlp_kernel<<<BB, HIDN, 0, stream>>>(scores, w1, b1, w2, b2, (float*)d_out);
}